// PhongCircleRenderer_76879914598772
// MI455X (gfx1250) — compile-verified
//
#include <hip/hip_runtime.h>
#include <math.h>

// ---------------------------------------------------------------------------
// PhongCircleRenderer for MI455X (gfx1250, wave32).
//
// Stage 1: Blinn-Phong shade 1.2M points  -> shaded[P] in d_ws (L2-resident).
// Stage 2: composite. Because alphas are exactly {0,1}, the pytorch3d-style
//          cumprod compositor collapses *exactly* to:
//              out = (idx0 < 0) ? background : shaded[idx0]
// Memory-bound (~160MB total); streams use non-temporal TH hints so the
// shaded gather buffer stays resident in the 192MB L2.
// ---------------------------------------------------------------------------

namespace {

constexpr float kAmbient  = 0.3f;
constexpr float kDiffuse  = 0.7f;
constexpr float kSpecular = 0.2f;

template <bool VEC4>
__global__ void __launch_bounds__(256)
shade_points_kernel(const float* __restrict__ points,
                    const float* __restrict__ features,
                    const float* __restrict__ normals,
                    const float* __restrict__ cam_centers,
                    const int*   __restrict__ cloud_idx,
                    const float* __restrict__ light_dir,
                    float*       __restrict__ shaded,   // [P,4] (VEC4) or [P,3]
                    int P)
{
    // Uniform: normalized light direction (scalar-path loads, tiny).
    const float lx0 = light_dir[0], ly0 = light_dir[1], lz0 = light_dir[2];
    const float linv = 1.0f / fmaxf(sqrtf(lx0*lx0 + ly0*ly0 + lz0*lz0), 1e-12f);
    const float lx = lx0 * linv, ly = ly0 * linv, lz = lz0 * linv;

    int i = blockIdx.x * blockDim.x + threadIdx.x;
    const int stride = gridDim.x * blockDim.x;
    for (; i < P; i += stride) {
        const size_t b3 = (size_t)i * 3;
        // One-shot streams: non-temporal so they don't evict the L2-resident
        // shaded buffer used by the gather stage.
        const float px = __builtin_nontemporal_load(points   + b3 + 0);
        const float py = __builtin_nontemporal_load(points   + b3 + 1);
        const float pz = __builtin_nontemporal_load(points   + b3 + 2);
        const float nx = __builtin_nontemporal_load(normals  + b3 + 0);
        const float ny = __builtin_nontemporal_load(normals  + b3 + 1);
        const float nz = __builtin_nontemporal_load(normals  + b3 + 2);
        const float fx = __builtin_nontemporal_load(features + b3 + 0);
        const float fy = __builtin_nontemporal_load(features + b3 + 1);
        const float fz = __builtin_nontemporal_load(features + b3 + 2);
        const int   c  = __builtin_nontemporal_load(cloud_idx + i);

        const float cx = cam_centers[3*c + 0];
        const float cy = cam_centers[3*c + 1];
        const float cz = cam_centers[3*c + 2];

        // view_dir = normalize(cam - p)   (v / max(||v||, 1e-12), as reference)
        float vx = cx - px, vy = cy - py, vz = cz - pz;
        const float vinv = 1.0f / fmaxf(sqrtf(vx*vx + vy*vy + vz*vz), 1e-12f);
        vx *= vinv; vy *= vinv; vz *= vinv;

        // half_vec = normalize(l + view_dir)
        float hx = lx + vx, hy = ly + vy, hz = lz + vz;
        const float hinv = 1.0f / fmaxf(sqrtf(hx*hx + hy*hy + hz*hz), 1e-12f);
        hx *= hinv; hy *= hinv; hz *= hinv;

        const float ndl = fmaxf(nx*lx + ny*ly + nz*lz, 0.0f);
        const float ndh = fmaxf(nx*hx + ny*hy + nz*hz, 0.0f);

        // ndh^32 via 5 exact squarings
        float s = ndh * ndh;  // ^2
        s *= s;               // ^4
        s *= s;               // ^8
        s *= s;               // ^16
        s *= s;               // ^32

        const float scale = kAmbient + kDiffuse * ndl;
        const float spec  = kSpecular * s;
        const float r = fminf(fmaxf(fx * scale + spec, 0.0f), 1.0f);
        const float g = fminf(fmaxf(fy * scale + spec, 0.0f), 1.0f);
        const float b = fminf(fmaxf(fz * scale + spec, 0.0f), 1.0f);

        if constexpr (VEC4) {
            float4 o; o.x = r; o.y = g; o.z = b; o.w = 0.0f;
            reinterpret_cast<float4*>(shaded)[i] = o;   // regular TH: keep in L2
        } else {
            shaded[b3 + 0] = r;
            shaded[b3 + 1] = g;
            shaded[b3 + 2] = b;
        }
    }
}

template <bool VEC4>
__global__ void __launch_bounds__(256)
composite_kernel(const int*   __restrict__ idx,     // [npix, K]
                 const float* __restrict__ shaded,  // [P,4] or [P,3]
                 float*       __restrict__ out,     // [npix, 3]
                 int npix, int K)
{
    int i = blockIdx.x * blockDim.x + threadIdx.x;
    const int stride = gridDim.x * blockDim.x;
    for (; i < npix; i += stride) {
        // gfx1250 prefetch of next grid-stride iteration's idx line
        const int nxt = i + stride;
        if (nxt < npix)
            __builtin_prefetch(idx + (size_t)nxt * K, 0, 0);

        // Exact collapse of the binary-alpha compositor: only fragment 0
        // matters (transmittance is 0 after the first valid hit, and pixels
        // whose first fragment is empty are overwritten with background).
        const int id0 = __builtin_nontemporal_load(idx + (size_t)i * K);

        float r, g, b;
        if (id0 < 0) {
            r = 1.0f; g = 1.0f; b = 1.0f;          // BG_COLOR
        } else if constexpr (VEC4) {
            const float4 s = reinterpret_cast<const float4*>(shaded)[id0];
            r = s.x; g = s.y; b = s.z;
        } else {
            const size_t sb = (size_t)id0 * 3;
            r = shaded[sb + 0]; g = shaded[sb + 1]; b = shaded[sb + 2];
        }

        const size_t o = (size_t)i * 3;
        __builtin_nontemporal_store(r, out + o + 0);
        __builtin_nontemporal_store(g, out + o + 1);
        __builtin_nontemporal_store(b, out + o + 2);
    }
}

} // namespace

extern "C" void kernel_launch(void* const* d_in, const int* in_sizes, int n_in,
                              void* d_out, int out_size, void* d_ws, size_t ws_size,
                              hipStream_t stream) {
    // setup_inputs() order:
    const int*   idx       = (const int*)  d_in[0];  // [B,H,W,K]
    const float* points    = (const float*)d_in[1];  // [P,3]
    const float* features  = (const float*)d_in[2];  // [P,3]
    const float* normals   = (const float*)d_in[3];  // [P,3]
    const float* cam       = (const float*)d_in[4];  // [B,3]
    const int*   cloud_idx = (const int*)  d_in[5];  // [P]
    const float* light     = (const float*)d_in[6];  // [3]
    float* out = (float*)d_out;

    const int P    = in_sizes[5];
    const int npix = out_size / 3;             // B*H*W
    const int K    = in_sizes[0] / npix;       // fragments per pixel (8)

    float* shaded = (float*)d_ws;
    const bool vec4 = ws_size >= (size_t)P * 16;  // padded float4 layout fits?

    const dim3 blk(256);                       // 8 wave32 per block
    const dim3 g_shade((P + 255) / 256);
    const dim3 g_comp(2048);                   // grid-stride, ~4 iters/thread

    if (vec4) {
        shade_points_kernel<true><<<g_shade, blk, 0, stream>>>(
            points, features, normals, cam, cloud_idx, light, shaded, P);
        composite_kernel<true><<<g_comp, blk, 0, stream>>>(
            idx, shaded, out, npix, K);
    } else {
        shade_points_kernel<false><<<g_shade, blk, 0, stream>>>(
            points, features, normals, cam, cloud_idx, light, shaded, P);
        composite_kernel<false><<<g_comp, blk, 0, stream>>>(
            idx, shaded, out, npix, K);
    }
}